// DecoderXL_47382079210071
// MI455X (gfx1250) — compile-verified
//
#include <hip/hip_runtime.h>
#include <hip/hip_bf16.h>

typedef __attribute__((ext_vector_type(2))) float v2f;
typedef __attribute__((ext_vector_type(4))) float v4f;
typedef __attribute__((ext_vector_type(8))) float v8f;

#define NL    4
#define NH    8
#define DM    512
#define DH    64
#define DI    2048
#define QLEN  1024
#define MLEN  1024
#define BATCH 4
#define KLEN  (QLEN + MLEN)

// Exact fp32 WMMA: D(16x16) = A(16x4) * B(4x16) + C
// A: M = lane&15; lanes 0-15 hold K={k,k+1}, lanes 16-31 hold K={k+2,k+3}
// B: N = lane&15; same K split -> both operands are contiguous K-pairs (b64)
// C/D: VGPR v -> row v (lanes 0-15) / row v+8 (lanes 16-31), col = lane&15
__device__ __forceinline__ v8f wmma_f32_16x16x4(v2f a, v2f b, v8f c) {
  return __builtin_amdgcn_wmma_f32_16x16x4_f32(false, a, false, b, (short)0, c,
                                               false, false);
}

// ---------------------------------------------------------------------------
// Generic NT GEMM: C[M,N] = A[M,K] * W[N,K]^T (+bias[n]) (+relu)
// Block: 256 threads (8 waves), tile 64(M) x 64(N), K-chunk 32.
// Wave (mw,nw): two 16x16 output tiles at rows mw*16 and mw*16+32, col nw*16,
// sharing one B fragment per WMMA step. All LDS strides 16B-aligned:
// staging uses b128, fragments use single b64 loads (no packing movs).
// ---------------------------------------------------------------------------
template <bool RELU, bool BIAS>
__global__ void __launch_bounds__(256) gemm_nt_wmma(
    const float* __restrict__ A, const float* __restrict__ W,
    const float* __restrict__ bias, float* __restrict__ C,
    int M, int N, int K) {
  __shared__ float As[64][36];
  __shared__ float Ws[64][36];
  const int m0 = blockIdx.y * 64;
  const int n0 = blockIdx.x * 64;
  const int tid = threadIdx.x;
  const int wave = tid >> 5, lane = tid & 31;
  const int hf = lane >> 4, lm = lane & 15;
  const int msub = (wave >> 2) * 16;  // 0 / 16 (+32 for second tile)
  const int nsub = (wave & 3) * 16;   // 0 / 16 / 32 / 48
  v8f acc0 = {}, acc1 = {};
  for (int k0 = 0; k0 < K; k0 += 32) {
    __syncthreads();
    for (int e = tid; e < 64 * 8; e += 256) {  // 512 x float4 per operand
      int r = e >> 3, c = (e & 7) << 2;
      *(v4f*)&As[r][c] = *(const v4f*)&A[(size_t)(m0 + r) * K + k0 + c];
      *(v4f*)&Ws[r][c] = *(const v4f*)&W[(size_t)(n0 + r) * K + k0 + c];
    }
    if (k0 + 32 < K) {  // speculative prefetch of next K-slab
      __builtin_prefetch(A + (size_t)(m0 + (tid & 63)) * K + k0 + 32, 0, 1);
      __builtin_prefetch(W + (size_t)(n0 + (tid & 63)) * K + k0 + 32, 0, 1);
    }
    __syncthreads();
#pragma unroll
    for (int kk = 0; kk < 32; kk += 4) {
      const int ko = kk + 2 * hf;
      const v2f b  = *(const v2f*)&Ws[nsub + lm][ko];
      const v2f a0 = *(const v2f*)&As[msub + lm][ko];
      const v2f a1 = *(const v2f*)&As[msub + 32 + lm][ko];
      acc0 = wmma_f32_16x16x4(a0, b, acc0);
      acc1 = wmma_f32_16x16x4(a1, b, acc1);
    }
  }
  const int col = n0 + nsub + lm;
  const float bv = BIAS ? bias[col] : 0.0f;
#pragma unroll
  for (int v = 0; v < 8; ++v) {
    int row = m0 + msub + v + (hf ? 8 : 0);
    float val0 = acc0[v] + bv;
    float val1 = acc1[v] + bv;
    if (RELU) { val0 = fmaxf(val0, 0.0f); val1 = fmaxf(val1, 0.0f); }
    C[(size_t)row * N + col] = val0;
    C[(size_t)(row + 32) * N + col] = val1;
  }
}

// ---------------------------------------------------------------------------
// Fused TXL relative attention (flash-style streaming softmax).
// grid = (QLEN/16, NH, BATCH); block = 256 (8 waves).
// Per 64-key chunk: AC = (Q+rwb)K^T (4 tiles), BDraw = (Q+rrb)R^T over an
// 80-wide rk window (5 tiles); combine with rel-shift gather jj-ii+15,
// causal mask, online softmax; waves 0-3 hold the 16x64 output accumulator.
// K buffer is reused for V stored TRANSPOSED so the P*V B-fragment is a
// contiguous b64 load as well.
// ---------------------------------------------------------------------------
#define PW 80
__global__ void __launch_bounds__(256) txl_attn(
    const float* __restrict__ wh,   // [KLEN, B, 3*NH*DH] (q|k|v)
    const float* __restrict__ rk,   // [KLEN, NH*DH]
    const float* __restrict__ rwb,  // [NH*DH]
    const float* __restrict__ rrb,  // [NH*DH]
    float* __restrict__ av) {       // [QLEN, B, NH*DH]
  __shared__ float Qw[16][DH + 4];
  __shared__ float Qr[16][DH + 4];
  __shared__ float KV[64][DH + 4];   // K rows [j][d], then V transposed [d][j]
  __shared__ float Rs[PW][DH + 4];
  __shared__ float Sac[16][64];
  __shared__ float Sbd[16][PW];
  __shared__ float P[16][DH + 4];
  __shared__ float red[16][17];
  __shared__ float mrow[16], lrow[16], cf[16];

  const int i0 = blockIdx.x * 16;
  const int n = blockIdx.y;
  const int b = blockIdx.z;
  const int tid = threadIdx.x;
  const int wave = tid >> 5, lane = tid & 31;
  const int hf = lane >> 4, lm = lane & 15;
  const float scale = 0.125f;  // 1/sqrt(DH)

  for (int e = tid; e < 16 * 16; e += 256) {  // Q tile + biases, vectorized
    int i = e >> 4, c = (e & 15) << 2;
    v4f q = *(const v4f*)&wh[((size_t)(MLEN + i0 + i) * BATCH + b) *
                                 (3 * NH * DH) + n * DH + c];
    v4f w = *(const v4f*)&rwb[n * DH + c];
    v4f r2 = *(const v4f*)&rrb[n * DH + c];
    *(v4f*)&Qw[i][c] = q + w;
    *(v4f*)&Qr[i][c] = q + r2;
  }
  if (tid < 16) { mrow[tid] = -3.0e38f; lrow[tid] = 0.0f; }
  v8f acc = {};

  const int nchunk = (i0 + 16 + MLEN + 63) >> 6;  // causal upper bound
  for (int c = 0; c < nchunk; ++c) {
    const int j0 = c * 64;
    const int pbase = j0 - i0 + (QLEN - 16);
    __syncthreads();
    for (int e = tid; e < 64 * 16; e += 256) {  // K chunk (row-major)
      int r = e >> 4, d = (e & 15) << 2;
      *(v4f*)&KV[r][d] = *(const v4f*)&wh[((size_t)(j0 + r) * BATCH + b) *
                                              (3 * NH * DH) + NH * DH +
                                          n * DH + d];
    }
    for (int e = tid; e < PW * 16; e += 256) {  // rk window (clamped)
      int r = e >> 4, d = (e & 15) << 2;
      int p = pbase + r;
      v4f val = {};
      if (p >= 0 && p < KLEN)
        val = *(const v4f*)&rk[(size_t)p * (NH * DH) + n * DH + d];
      *(v4f*)&Rs[r][d] = val;
    }
    __syncthreads();
    // 9 score tiles: t<4 -> AC columns, t>=4 -> BD columns
    for (int t = wave; t < 9; t += 8) {
      v8f s = {};
      if (t < 4) {
#pragma unroll
        for (int kk = 0; kk < DH; kk += 4) {
          const int ko = kk + 2 * hf;
          const v2f a = *(const v2f*)&Qw[lm][ko];
          const v2f bb = *(const v2f*)&KV[t * 16 + lm][ko];
          s = wmma_f32_16x16x4(a, bb, s);
        }
#pragma unroll
        for (int v = 0; v < 8; ++v) Sac[v + (hf ? 8 : 0)][t * 16 + lm] = s[v];
      } else {
        const int tt = t - 4;
#pragma unroll
        for (int kk = 0; kk < DH; kk += 4) {
          const int ko = kk + 2 * hf;
          const v2f a = *(const v2f*)&Qr[lm][ko];
          const v2f bb = *(const v2f*)&Rs[tt * 16 + lm][ko];
          s = wmma_f32_16x16x4(a, bb, s);
        }
#pragma unroll
        for (int v = 0; v < 8; ++v) Sbd[v + (hf ? 8 : 0)][tt * 16 + lm] = s[v];
      }
    }
    __syncthreads();
    // reuse KV buffer: stage V TRANSPOSED while stats run
    for (int e = tid; e < 64 * 16; e += 256) {
      int r = e >> 4, d = (e & 15) << 2;
      v4f t = *(const v4f*)&wh[((size_t)(j0 + r) * BATCH + b) * (3 * NH * DH) +
                               2 * NH * DH + n * DH + d];
      KV[d + 0][r] = t.x;
      KV[d + 1][r] = t.y;
      KV[d + 2][r] = t.z;
      KV[d + 3][r] = t.w;
    }
    // combine AC + rel-shifted BD, causal mask, online softmax stats
    {
      const int i = tid >> 4, l16 = tid & 15;
      float mx = -3.0e38f;
#pragma unroll
      for (int q4 = 0; q4 < 4; ++q4) {
        int jj = l16 + q4 * 16;
        float sv = (j0 + jj > i0 + i + MLEN)
                       ? -3.0e30f
                       : scale * (Sac[i][jj] + Sbd[i][jj - i + 15]);
        P[i][jj] = sv;
        mx = fmaxf(mx, sv);
      }
      red[i][l16] = mx;
      __syncthreads();
      if (tid < 16) {
        float m2 = red[tid][0];
#pragma unroll
        for (int t = 1; t < 16; ++t) m2 = fmaxf(m2, red[tid][t]);
        float mnew = fmaxf(mrow[tid], m2);
        cf[tid] = __expf(mrow[tid] - mnew);
        mrow[tid] = mnew;
      }
      __syncthreads();
      float ls = 0.0f;
      const float mi = mrow[i];
#pragma unroll
      for (int q4 = 0; q4 < 4; ++q4) {
        int jj = l16 + q4 * 16;
        float ev = __expf(P[i][jj] - mi);
        P[i][jj] = ev;
        ls += ev;
      }
      red[i][l16] = ls;
      __syncthreads();
      if (tid < 16) {
        float s2 = 0.0f;
#pragma unroll
        for (int t = 0; t < 16; ++t) s2 += red[tid][t];
        lrow[tid] = lrow[tid] * cf[tid] + s2;
      }
      __syncthreads();
    }
    // O += P * V^T-stored  (waves 0-3 own 16-wide dh sub-tiles)
    if (wave < 4) {
#pragma unroll
      for (int v = 0; v < 8; ++v) acc[v] *= cf[v + (hf ? 8 : 0)];
#pragma unroll
      for (int kk = 0; kk < 64; kk += 4) {
        const int ko = kk + 2 * hf;
        const v2f a = *(const v2f*)&P[lm][ko];
        const v2f bb = *(const v2f*)&KV[wave * 16 + lm][ko];
        acc = wmma_f32_16x16x4(a, bb, acc);
      }
    }
  }
  if (wave < 4) {
#pragma unroll
    for (int v = 0; v < 8; ++v) {
      int i = v + (hf ? 8 : 0);
      float o = acc[v] / lrow[i];
      av[((size_t)(i0 + i) * BATCH + b) * (NH * DH) + n * DH + wave * 16 + lm] =
          o;
    }
  }
}

// ---------------------------------------------------------------------------
// h = LayerNorm(h + a) * s + b   (one block per row of DM=512)
// ---------------------------------------------------------------------------
__global__ void __launch_bounds__(256) add_ln_kernel(
    float* __restrict__ h, const float* __restrict__ a,
    const float* __restrict__ sc, const float* __restrict__ bi) {
  __shared__ float red[256];
  const int row = blockIdx.x;
  const int tid = threadIdx.x;
  const size_t base = (size_t)row * DM;
  float x0 = h[base + tid] + a[base + tid];
  float x1 = h[base + tid + 256] + a[base + tid + 256];
  red[tid] = x0 + x1;
  __syncthreads();
  for (int off = 128; off > 0; off >>= 1) {
    if (tid < off) red[tid] += red[tid + off];
    __syncthreads();
  }
  float mu = red[0] * (1.0f / DM);
  __syncthreads();
  float d0 = x0 - mu, d1 = x1 - mu;
  red[tid] = d0 * d0 + d1 * d1;
  __syncthreads();
  for (int off = 128; off > 0; off >>= 1) {
    if (tid < off) red[tid] += red[tid + off];
    __syncthreads();
  }
  float rstd = rsqrtf(red[0] * (1.0f / DM) + 1e-5f);
  h[base + tid] = d0 * rstd * sc[tid] + bi[tid];
  h[base + tid + 256] = d1 * rstd * sc[tid + 256] + bi[tid + 256];
}

// ---------------------------------------------------------------------------
extern "C" void kernel_launch(void* const* d_in, const int* in_sizes, int n_in,
                              void* d_out, int out_size, void* d_ws,
                              size_t ws_size, hipStream_t stream) {
  const float* dec_inp = (const float*)d_in[0];
  const float* pos_emb = (const float*)d_in[1];
  const float* mems = (const float*)d_in[2];
  const float* r_w_bias = (const float*)d_in[3];
  const float* r_r_bias = (const float*)d_in[4];
  const float* qkv_w = (const float*)d_in[5];
  const float* qkv_b = (const float*)d_in[6];
  const float* r_w = (const float*)d_in[7];
  const float* o_w = (const float*)d_in[8];
  const float* ln1_s = (const float*)d_in[9];
  const float* ln1_b = (const float*)d_in[10];
  const float* ff_w1 = (const float*)d_in[11];
  const float* ff_b1 = (const float*)d_in[12];
  const float* ff_w2 = (const float*)d_in[13];
  const float* ff_b2 = (const float*)d_in[14];
  const float* ln2_s = (const float*)d_in[15];
  const float* ln2_b = (const float*)d_in[16];

  float* h = (float*)d_out;  // [QLEN, B, DM] — evolved in place
  char* ws = (char*)d_ws;
  size_t off = 0;
  float* catb = (float*)(ws + off);   off += (size_t)KLEN * BATCH * DM * 4;
  float* wheads = (float*)(ws + off); off += (size_t)KLEN * BATCH * 3 * NH * DH * 4;
  float* rkbuf = (float*)(ws + off);  off += (size_t)KLEN * NH * DH * 4;
  float* avbuf = (float*)(ws + off);  off += (size_t)QLEN * BATCH * NH * DH * 4;
  float* proj = (float*)(ws + off);   off += (size_t)QLEN * BATCH * DM * 4;
  float* ffbuf = (float*)(ws + off);  off += (size_t)QLEN * BATCH * DI * 4;

  hipMemcpyAsync(h, dec_inp, (size_t)QLEN * BATCH * DM * 4,
                 hipMemcpyDeviceToDevice, stream);

  const int Mqk = KLEN * BATCH;  // 8192
  const int Mq = QLEN * BATCH;   // 4096
  for (int l = 0; l < NL; ++l) {
    // cat = [mems[l]; h]  (layouts are flat-contiguous -> two d2d copies)
    hipMemcpyAsync(catb, mems + (size_t)l * MLEN * BATCH * DM,
                   (size_t)MLEN * BATCH * DM * 4, hipMemcpyDeviceToDevice,
                   stream);
    hipMemcpyAsync(catb + (size_t)MLEN * BATCH * DM, h,
                   (size_t)QLEN * BATCH * DM * 4, hipMemcpyDeviceToDevice,
                   stream);
    // w_heads = cat @ qkv_w^T + qkv_b
    gemm_nt_wmma<false, true>
        <<<dim3((3 * NH * DH) / 64, Mqk / 64), 256, 0, stream>>>(
            catb, qkv_w + (size_t)l * 3 * NH * DH * DM,
            qkv_b + (size_t)l * 3 * NH * DH, wheads, Mqk, 3 * NH * DH, DM);
    // rk = pos_emb @ r_w^T
    gemm_nt_wmma<false, false>
        <<<dim3((NH * DH) / 64, KLEN / 64), 256, 0, stream>>>(
            pos_emb, r_w + (size_t)l * NH * DH * DM, nullptr, rkbuf, KLEN,
            NH * DH, DM);
    // fused relative attention -> avbuf
    txl_attn<<<dim3(QLEN / 16, NH, BATCH), 256, 0, stream>>>(
        wheads, rkbuf, r_w_bias, r_r_bias, avbuf);
    // out projection
    gemm_nt_wmma<false, false><<<dim3(DM / 64, Mq / 64), 256, 0, stream>>>(
        avbuf, o_w + (size_t)l * DM * NH * DH, nullptr, proj, Mq, DM, NH * DH);
    add_ln_kernel<<<Mq, 256, 0, stream>>>(h, proj, ln1_s + l * DM,
                                          ln1_b + l * DM);
    // FFN
    gemm_nt_wmma<true, true><<<dim3(DI / 64, Mq / 64), 256, 0, stream>>>(
        h, ff_w1 + (size_t)l * DI * DM, ff_b1 + l * DI, ffbuf, Mq, DI, DM);
    gemm_nt_wmma<false, true><<<dim3(DM / 64, Mq / 64), 256, 0, stream>>>(
        ffbuf, ff_w2 + (size_t)l * DM * DI, ff_b2 + l * DM, proj, Mq, DM, DI);
    add_ln_kernel<<<Mq, 256, 0, stream>>>(h, proj, ln2_s + l * DM,
                                          ln2_b + l * DM);
  }
  (void)in_sizes; (void)n_in; (void)out_size; (void)ws_size;
}